// Renderer_N16_54846732369925
// MI455X (gfx1250) — compile-verified
//
#include <hip/hip_runtime.h>
#include <hip/hip_bf16.h>
#include <math.h>

typedef float v2f __attribute__((ext_vector_type(2)));
typedef float v8f __attribute__((ext_vector_type(8)));

#define KS   51
#define PAD  25
#define IMW  512
#define IMH  512
#define NCH  32   // 4 batches * 8 masks

__device__ __forceinline__ int iclampi(int x, int lo, int hi) {
    return x < lo ? lo : (x > hi ? hi : x);
}
__device__ __forceinline__ int reflect512(int g) {
    g = (g < 0) ? -g : g;
    g = (g >= IMW) ? (2 * IMW - 2 - g) : g;
    return g;
}

// ---------------- Stage 0: global mean of raw input (deterministic 2-stage) --
__global__ __launch_bounds__(256) void k_reduce_partial(const float* __restrict__ in,
                                                        float* __restrict__ partials,
                                                        int n) {
    __shared__ float sd[256];
    int tid = threadIdx.x;
    float s = 0.f;
    for (int i = blockIdx.x * 256 + tid; i < n; i += gridDim.x * 256) s += in[i];
    sd[tid] = s;
    __syncthreads();
    for (int st = 128; st > 0; st >>= 1) {
        if (tid < st) sd[tid] += sd[tid + st];
        __syncthreads();
    }
    if (tid == 0) partials[blockIdx.x] = sd[0];
}

__global__ __launch_bounds__(256) void k_reduce_final(const float* __restrict__ partials,
                                                      float* __restrict__ sumSlot) {
    __shared__ float sd[256];
    int tid = threadIdx.x;
    sd[tid] = partials[tid] + partials[tid + 256] + partials[tid + 512] + partials[tid + 768];
    __syncthreads();
    for (int st = 128; st > 0; st >>= 1) {
        if (tid < st) sd[tid] += sd[tid + st];
        __syncthreads();
    }
    if (tid == 0) sumSlot[0] = sd[0];
}

// ---------------- Stage 1: per-channel 1D Gaussian weights + grade coeffs ----
// wts:   [NCH][64]  (51 valid taps, zero padded)
// coefs: [NCH][10]  = {A0,A1,A2, G0,G1,G2, B0,B1,B2, C}
__global__ __launch_bounds__(64) void k_prep(const float* __restrict__ params,
                                             const float* __restrict__ sumSlot,
                                             float* __restrict__ wts,
                                             float* __restrict__ coefs) {
    const int ch = blockIdx.x;
    const float* p = params + ch * 17;
    const int tid = threadIdx.x;

    float alpha = p[16];
    alpha = fminf(fmaxf(alpha, 0.f), 1.f);
    const float sigma = alpha * (float)(KS - 3) + 3.f;     // MIN_SIGMA = 3
    const float denom = 6.2831852f * sigma * sigma;        // TWO_PI * sigma^2

    // r = linspace(-26, 25, 51): step 51/50 = 1.02
    const float ri  = -26.f + 1.02f * (float)tid;
    const float val = (tid < KS) ? expf(-(ri * ri) / denom) : 0.f;

    __shared__ float red[64];
    red[tid] = val;
    __syncthreads();
    for (int st = 32; st > 0; st >>= 1) {
        if (tid < st) red[tid] += red[tid + st];
        __syncthreads();
    }
    const float inv = 1.f / red[0];
    wts[ch * 64 + tid] = val * inv;

    if (tid == 0) {
        float mean = sumSlot[0] * (1.f / (float)(4 * 3 * IMH * IMW));
        mean = (mean + 1.f) * 0.5f;                        // mean of (input+1)/2

        const float contrast = p[0] + 1.f;
        const float sat      = (p[1] + 1.f) * 0.5f;
        const float hue      = (p[2] + 1.f) * 0.5f;
        const float temp     = (p[3] + 1.f) * 0.5f;
        const float t        = 2.f * temp - 1.f;

        float hf[3] = { (hue - 0.5f) * 0.2f, (0.5f - hue) * 0.1f, (hue - 0.5f) * 0.2f };
        float x0 = t * (125.f / 128.f);
        float x1 = t * (-29.f / 128.f);
        float x2 = t * (-112.f / 128.f);
        float tf[3];
        tf[0] = (x0 >= 0.f) ? x0 : (24.f / 125.f) * x0;
        tf[1] = (x1 >= 0.f) ? x1 : 0.f;
        tf[2] = (x2 >= 0.f) ? x2 : (52.f / 112.f) * x2;

        const float baseB = (1.f - contrast) * mean;
        float* cf = coefs + ch * 10;
        #pragma unroll
        for (int c = 0; c < 3; ++c) {
            const float lift  = p[4 + c];
            const float gamma = p[7 + c];
            const float gain  = p[10 + c];
            const float shift = p[13 + c];
            cf[c]     = contrast - 2.f + 2.f * sat + hf[c] + tf[c] - 2.f * lift + 2.f * gain;
            cf[3 + c] = 7.2f * gamma;
            cf[6 + c] = baseB + 2.f * lift + shift;
        }
        cf[9] = 1.f - 2.f * sat;
    }
}

// ---------------- Stage 2: horizontal 51-tap blur via V_WMMA_F32_16X16X4_F32
// Block: 256 threads = 8 waves; covers 16 rows x 128 cols of one channel.
// Wave w computes a 16x16 tile:  D(16x16) += A(16x4 image rows) x B(4x16 band).
// Band fragments hoisted to registers; A is one ds_load_b64 per step; the
// 17-step K loop is fully unrolled -> inner body is ds_load_b64 + v_wmma.
#define HPITCH 186   // even (8B-aligned v2f loads), stride mod 64 conflict-free
__global__ __launch_bounds__(256) void k_hblur(const float* __restrict__ masks,
                                               const float* __restrict__ wts,
                                               float* __restrict__ tmp) {
    const int ch   = blockIdx.z;
    const int row0 = blockIdx.y * 16;
    const int col0 = blockIdx.x * 128;

    __shared__ float patch[16 * HPITCH];    // cols [0,178) data, [178,186) zero
    __shared__ float wl[64];

    const int tid = threadIdx.x;
    if (tid < 64) wl[tid] = wts[ch * 64 + tid];   // taps >= 51 are zero

    const float* src = masks + (size_t)ch * (IMH * IMW) + (size_t)row0 * IMW;
    for (int idx = tid; idx < 16 * HPITCH; idx += 256) {
        const int r = idx / HPITCH;
        const int c = idx - r * HPITCH;
        float v = 0.f;
        if (c < 178) v = src[r * IMW + reflect512(col0 - PAD + c)];
        patch[idx] = v;
    }
    __syncthreads();

    const int lane = tid & 31;
    const int wv   = tid >> 5;               // wave = which 16-col tile
    const int n    = lane & 15;
    const int off  = (lane < 16) ? 0 : 2;    // K-pair this half-wave owns

    // Hoisted band fragments: B(t, n) = w[t - n], t = 4j + off (+1)
    float bx[17], by[17];
    #pragma unroll
    for (int j = 0; j < 17; ++j) {
        const int k0 = 4 * j + off - n;
        const int k1 = k0 + 1;
        float v0 = wl[iclampi(k0, 0, 50)];
        float v1 = wl[iclampi(k1, 0, 50)];
        bx[j] = ((unsigned)k0 > 50u) ? 0.f : v0;
        by[j] = ((unsigned)k1 > 50u) ? 0.f : v1;
    }

    const float* arow = patch + n * HPITCH + 16 * wv + off;  // 8B aligned
    v8f acc = {};
    #pragma unroll
    for (int j = 0; j < 17; ++j) {
        v2f a = *(const v2f*)(arow + 4 * j);                 // ds_load_b64
        v2f b; b.x = bx[j]; b.y = by[j];
        acc = __builtin_amdgcn_wmma_f32_16x16x4_f32(false, a, false, b,
                                                    (short)0, acc, false, false);
    }

    float* dst = tmp + (size_t)ch * (IMH * IMW) + (size_t)row0 * IMW + col0 + 16 * wv;
    const int rsel = (lane < 16) ? 0 : 8;
    #pragma unroll
    for (int v = 0; v < 8; ++v) dst[(v + rsel) * IMW + n] = acc[v];
}

// ---------------- Stage 3: vertical blur (WMMA) fused with color grading ----
// Block: 256 threads = 8 waves; covers 128 rows x 16 cols of one batch.
// Reflected-row offsets hoisted (shared across all 8 masks); band fragments
// hoisted per mask; 17-step K loop unrolled -> 2 global loads + v_wmma.
__global__ __launch_bounds__(256) void k_vblur_grade(const float* __restrict__ input,
                                                     const float* __restrict__ tmp,
                                                     const float* __restrict__ wts,
                                                     const float* __restrict__ coefs,
                                                     float* __restrict__ out) {
    const int b     = blockIdx.z;
    const int col0  = blockIdx.x * 16;
    const int rows0 = blockIdx.y * 128;

    __shared__ float wl[8][64];
    const int tid = threadIdx.x;
    for (int idx = tid; idx < 8 * 64; idx += 256) {
        const int m = idx >> 6, k = idx & 63;
        wl[m][k] = wts[(b * 8 + m) * 64 + k];
    }
    __syncthreads();

    const int lane = tid & 31;
    const int wv   = tid >> 5;
    const int n    = lane & 15;              // A row (M) and B col (N)
    const int off  = (lane < 16) ? 0 : 2;
    const int tileRow = rows0 + wv * 16;
    const int t0      = tileRow - PAD;

    // Hoisted reflected-row element offsets (mask-independent)
    int rofs[34];
    #pragma unroll
    for (int j = 0; j < 17; ++j) {
        const int t = 4 * j + off;
        rofs[2 * j]     = reflect512(t0 + t)     * IMW + col0 + n;
        rofs[2 * j + 1] = reflect512(t0 + t + 1) * IMW + col0 + n;
    }

    float sA0[8] = {}, sA1[8] = {}, sA2[8] = {};
    float sG0[8] = {}, sG1[8] = {}, sG2[8] = {};
    float sB0[8] = {}, sB1[8] = {}, sB2[8] = {};
    float sC[8]  = {};

    #pragma unroll 1
    for (int m = 0; m < 8; ++m) {
        const float* tch = tmp + (size_t)(b * 8 + m) * (IMH * IMW);

        // Band fragments for this mask: A(r, t) = w[t - r]
        float ax[17], ay[17];
        #pragma unroll
        for (int j = 0; j < 17; ++j) {
            const int k0 = 4 * j + off - n;
            const int k1 = k0 + 1;
            float v0 = wl[m][iclampi(k0, 0, 50)];
            float v1 = wl[m][iclampi(k1, 0, 50)];
            ax[j] = ((unsigned)k0 > 50u) ? 0.f : v0;
            ay[j] = ((unsigned)k1 > 50u) ? 0.f : v1;
        }

        v8f acc = {};
        #pragma unroll
        for (int j = 0; j < 17; ++j) {
            v2f a;  a.x  = ax[j];             a.y  = ay[j];
            v2f bb; bb.x = tch[rofs[2 * j]];  bb.y = tch[rofs[2 * j + 1]];
            acc = __builtin_amdgcn_wmma_f32_16x16x4_f32(false, a, false, bb,
                                                        (short)0, acc, false, false);
        }

        const float* cf = coefs + (b * 8 + m) * 10;   // uniform -> scalar loads
        const float cA0 = cf[0], cA1 = cf[1], cA2 = cf[2];
        const float cG0 = cf[3], cG1 = cf[4], cG2 = cf[5];
        const float cB0 = cf[6], cB1 = cf[7], cB2 = cf[8];
        const float cCC = cf[9];
        #pragma unroll
        for (int v = 0; v < 8; ++v) {
            const float bl = acc[v];
            sA0[v] += cA0 * bl; sA1[v] += cA1 * bl; sA2[v] += cA2 * bl;
            sG0[v] += cG0 * bl; sG1[v] += cG1 * bl; sG2[v] += cG2 * bl;
            sB0[v] += cB0 * bl; sB1[v] += cB1 * bl; sB2[v] += cB2 * bl;
            sC[v]  += cCC * bl;
        }
    }

    // Pointwise grade: out = in + 2*(img*SA + L*SC + q*SG + SB) per channel
    const size_t plane = (size_t)IMH * IMW;
    const float* inb = input + (size_t)b * 3 * plane;
    float*      outb = out   + (size_t)b * 3 * plane;
    const int rsel = (lane < 16) ? 0 : 8;

    #pragma unroll
    for (int v = 0; v < 8; ++v) {
        const int row  = tileRow + v + rsel;
        const size_t pix = (size_t)row * IMW + col0 + n;
        const float i0 = inb[pix], i1 = inb[plane + pix], i2 = inb[2 * plane + pix];
        const float g0 = (i0 + 1.f) * 0.5f;
        const float g1 = (i1 + 1.f) * 0.5f;
        const float g2 = (i2 + 1.f) * 0.5f;
        // sRGB -> linear -> Lab L / 100
        const float l0 = (g0 > 0.04045f) ? powf((g0 + 0.055f) * (1.f / 1.055f), 2.4f) : g0 * (1.f / 12.92f);
        const float l1 = (g1 > 0.04045f) ? powf((g1 + 0.055f) * (1.f / 1.055f), 2.4f) : g1 * (1.f / 12.92f);
        const float l2 = (g2 > 0.04045f) ? powf((g2 + 0.055f) * (1.f / 1.055f), 2.4f) : g2 * (1.f / 12.92f);
        const float y  = 0.212671f * l0 + 0.71516f * l1 + 0.072169f * l2;
        float L = (y > 0.008856f) ? (116.f * cbrtf(y) - 16.f) : 903.3f * y;
        L *= 0.01f;
        const float q0 = 0.25f - (g0 - 0.5f) * (g0 - 0.5f);
        const float q1 = 0.25f - (g1 - 0.5f) * (g1 - 0.5f);
        const float q2 = 0.25f - (g2 - 0.5f) * (g2 - 0.5f);
        outb[pix]             = i0 + 2.f * (g0 * sA0[v] + L * sC[v] + q0 * sG0[v] + sB0[v]);
        outb[plane + pix]     = i1 + 2.f * (g1 * sA1[v] + L * sC[v] + q1 * sG1[v] + sB1[v]);
        outb[2 * plane + pix] = i2 + 2.f * (g2 * sA2[v] + L * sC[v] + q2 * sG2[v] + sB2[v]);
    }
}

// ---------------------------------------------------------------------------
extern "C" void kernel_launch(void* const* d_in, const int* in_sizes, int n_in,
                              void* d_out, int out_size, void* d_ws, size_t ws_size,
                              hipStream_t stream) {
    const float* input  = (const float*)d_in[0];   // (4,3,512,512)
    const float* masks  = (const float*)d_in[1];   // (4,8,512,512)
    const float* params = (const float*)d_in[2];   // (4,8,1,17)
    float* out = (float*)d_out;

    float* ws       = (float*)d_ws;
    float* partials = ws;               // [1024]
    float* sumSlot  = ws + 1024;        // [1]
    float* coefs    = ws + 1056;        // [32][10]
    float* wts      = ws + 1536;        // [32][64]
    float* tmp      = ws + 4096;        // [32][512][512] horizontal-blur result

    const int nIn = 4 * 3 * IMH * IMW;

    k_reduce_partial<<<1024, 256, 0, stream>>>(input, partials, nIn);
    k_reduce_final<<<1, 256, 0, stream>>>(partials, sumSlot);
    k_prep<<<32, 64, 0, stream>>>(params, sumSlot, wts, coefs);

    dim3 g2(IMW / 128, IMH / 16, NCH);          // (4, 32, 32)
    k_hblur<<<g2, 256, 0, stream>>>(masks, wts, tmp);

    dim3 g3(IMW / 16, IMH / 128, 4);            // (32, 4, 4)
    k_vblur_grade<<<g3, 256, 0, stream>>>(input, tmp, wts, coefs, out);
}